// ProjChamfer_test_35115652612620
// MI455X (gfx1250) — compile-verified
//
#include <hip/hip_runtime.h>

// ProjChamfer on MI455X (gfx1250):
//   d[n,m] = |a_n|^2 + |b_m|^2 - 2 a.b  ==  [ax,ay,|a|^2,1] . [-2bx,-2by,1,|b|^2]
// -> one V_WMMA_F32_16X16X4_F32 per 16x16 distance tile (K=4 packs the bias terms).
// Round-3 changes vs round-2:
//  * eps-clamp via v_med3_num_f32 (no fminf/fmaxf canonicalization churn)
//  * entire min-reduction tree in u32 bit-pattern space (positive floats order
//    like unsigned ints) -> v_min_u32/v_min3_u32, zero canonicalize ops
//  * row-min accumulators are uint; converted once at the end

typedef __attribute__((ext_vector_type(2))) float v2f;
typedef __attribute__((ext_vector_type(8))) float v8f;

#define BB 8
#define NN 4096
#define BN (BB * NN)
#define NT (NN / 16)          // 256 column tiles
#define CENTER 112.0f
#define INF_BITS 0x7F800000u

__device__ __forceinline__ unsigned umin2(unsigned a, unsigned b) {
    return a < b ? a : b;      // v_min_u32
}
__device__ __forceinline__ unsigned clamp_bits(float d) {
    // max(d, 1e-12) as a single v_med3_num_f32, then bit-cast
    return __float_as_uint(__builtin_amdgcn_fmed3f(d, 1e-12f, 3.0e38f));
}

// ---------------- Kernel 1: project, normalize, pack WMMA operands ----------
__global__ void pc_pack_kernel(const float* __restrict__ pred,
                               const float* __restrict__ gt,
                               const float* __restrict__ P,
                               float* __restrict__ Apk,
                               float* __restrict__ Bpk,
                               unsigned* __restrict__ gCol) {
    int t = blockIdx.x * blockDim.x + threadIdx.x;
    if (t >= BN) return;
    int b = t / NN;
    const float* Pm = P + b * 12;
    float p00 = Pm[0], p01 = Pm[1],  p02 = Pm[2],  p03 = Pm[3];
    float p10 = Pm[4], p11 = Pm[5],  p12 = Pm[6],  p13 = Pm[7];
    float p20 = Pm[8], p21 = Pm[9],  p22 = Pm[10], p23 = Pm[11];
    const float inv2c = 1.0f / (2.0f * CENTER);

    // gt -> A-format row [ax, ay, |a|^2, 1]
    {
        float x = gt[t * 3 + 0], y = gt[t * 3 + 1], z = gt[t * 3 + 2];
        float w  = p20 * x + p21 * y + p22 * z + p23;
        float iw = 1.0f / w;
        float ax = ((p00 * x + p01 * y + p02 * z + p03) * iw - CENTER) * inv2c;
        float ay = ((p10 * x + p11 * y + p12 * z + p13) * iw - CENTER) * inv2c;
        Apk[t * 4 + 0] = ax;
        Apk[t * 4 + 1] = ay;
        Apk[t * 4 + 2] = ax * ax + ay * ay;
        Apk[t * 4 + 3] = 1.0f;
    }
    // pred -> B-format col [-2bx, -2by, 1, |b|^2]
    {
        float x = pred[t * 3 + 0], y = pred[t * 3 + 1], z = pred[t * 3 + 2];
        float w  = p20 * x + p21 * y + p22 * z + p23;
        float iw = 1.0f / w;
        float bx = ((p00 * x + p01 * y + p02 * z + p03) * iw - CENTER) * inv2c;
        float by = ((p10 * x + p11 * y + p12 * z + p13) * iw - CENTER) * inv2c;
        Bpk[t * 4 + 0] = -2.0f * bx;
        Bpk[t * 4 + 1] = -2.0f * by;
        Bpk[t * 4 + 2] = 1.0f;
        Bpk[t * 4 + 3] = bx * bx + by * by;
    }
    gCol[t] = INF_BITS;
}

// ---------------- Kernel 2: WMMA-tiled chamfer mins -------------------------
// Block = 256 threads = 8 waves. Each wave owns a 32-row (2 x 16) gt slab and
// sweeps all 256 pred column tiles. Grid = B * (N/256) = 128 blocks.
__global__ void pc_chamfer_kernel(const float* __restrict__ Apk,
                                  const float* __restrict__ Bpk,
                                  unsigned* __restrict__ gCol,
                                  float* __restrict__ gRow) {
    __shared__ unsigned sCol[NN];  // 16 KB of 320 KB WGP LDS

    const int b     = blockIdx.x >> 4;   // 16 row-chunks (of 256 rows) per batch
    const int chunk = blockIdx.x & 15;
    const int lane  = threadIdx.x & 31;
    const int wave  = threadIdx.x >> 5;
    const int lmod  = lane & 15;
    const int lhalf = lane >> 4;

    for (int i = threadIdx.x; i < NN; i += blockDim.x) sCol[i] = INF_BITS;
    __syncthreads();

    const int row0 = chunk * 256 + wave * 32;

    // A fragments: lane l, VGPR v -> A[row + l%16][(l/16)*2 + v]  (one b64 load each)
    const float* aptr0 = Apk + ((size_t)b * NN + row0 + lmod) * 4 + lhalf * 2;
    v2f afrag0 = *(const v2f*)aptr0;
    v2f afrag1 = *(const v2f*)(aptr0 + 64);  // rows row0+16 .. row0+31

    unsigned rmin0[8], rmin1[8];
#pragma unroll
    for (int r = 0; r < 8; ++r) { rmin0[r] = INF_BITS; rmin1[r] = INF_BITS; }

    const float* bbase = Bpk + ((size_t)b * NN + lmod) * 4 + lhalf * 2;

    // software pipeline: prefetch B tile j+1 while computing tile j
    v2f bcur = *(const v2f*)bbase;
    for (int j = 0; j < NT; ++j) {
        int jn = (j + 1 < NT) ? (j + 1) : j;
        v2f bnext = *(const v2f*)(bbase + (size_t)jn * 64);

        v8f z = (v8f)0.0f;
        // two independent v_wmma_f32_16x16x4_f32 sharing one B fragment
        v8f d0 = __builtin_amdgcn_wmma_f32_16x16x4_f32(
            false, afrag0, false, bcur, (short)0, z, false, false);
        v8f d1 = __builtin_amdgcn_wmma_f32_16x16x4_f32(
            false, afrag1, false, bcur, (short)0, z, false, false);

        unsigned cm = INF_BITS;
#pragma unroll
        for (int r = 0; r < 8; ++r) {
            unsigned u0 = clamp_bits(d0[r]);   // v_med3_num_f32
            unsigned u1 = clamp_bits(d1[r]);
            rmin0[r] = umin2(rmin0[r], u0);    // v_min_u32 / v_min3_u32
            rmin1[r] = umin2(rmin1[r], u1);
            cm       = umin2(cm, umin2(u0, u1));
        }
        // lanes l and l+16 hold the same column n=l%16 (over different rows);
        // both atomically min into the same slot -- no shuffle, no divergence.
        atomicMin(&sCol[j * 16 + lmod], cm);   // ds_min_u32

        bcur = bnext;
    }

    // finalize row mins: reduce across the 16 lanes of each half (u32 mins)
#pragma unroll
    for (int r = 0; r < 8; ++r) {
        unsigned rm0 = rmin0[r];
        rm0 = umin2(rm0, (unsigned)__shfl_xor((int)rm0, 1, 32));
        rm0 = umin2(rm0, (unsigned)__shfl_xor((int)rm0, 2, 32));
        rm0 = umin2(rm0, (unsigned)__shfl_xor((int)rm0, 4, 32));
        rm0 = umin2(rm0, (unsigned)__shfl_xor((int)rm0, 8, 32));
        unsigned rm1 = rmin1[r];
        rm1 = umin2(rm1, (unsigned)__shfl_xor((int)rm1, 1, 32));
        rm1 = umin2(rm1, (unsigned)__shfl_xor((int)rm1, 2, 32));
        rm1 = umin2(rm1, (unsigned)__shfl_xor((int)rm1, 4, 32));
        rm1 = umin2(rm1, (unsigned)__shfl_xor((int)rm1, 8, 32));
        if (lmod == 0) {
            gRow[(size_t)b * NN + row0 + lhalf * 8 + r]      = __uint_as_float(rm0);
            gRow[(size_t)b * NN + row0 + 16 + lhalf * 8 + r] = __uint_as_float(rm1);
        }
    }

    __syncthreads();
    for (int i = threadIdx.x; i < NN; i += blockDim.x)
        atomicMin(&gCol[(size_t)b * NN + i], sCol[i]);  // global_atomic_min_u32
}

// ---------------- Kernel 3: sqrt + means + combine --------------------------
__global__ void pc_reduce_kernel(const unsigned* __restrict__ gCol,
                                 const float* __restrict__ gRow,
                                 float* __restrict__ out) {
    __shared__ float s0s[256];
    __shared__ float s1s[256];
    float s0 = 0.0f, s1 = 0.0f;
    for (int i = threadIdx.x; i < BN; i += blockDim.x) {
        s0 += sqrtf(__uint_as_float(gCol[i]));
        s1 += sqrtf(gRow[i]);
    }
    s0s[threadIdx.x] = s0;
    s1s[threadIdx.x] = s1;
    __syncthreads();
    for (int s = 128; s > 0; s >>= 1) {
        if (threadIdx.x < s) {
            s0s[threadIdx.x] += s0s[threadIdx.x + s];
            s1s[threadIdx.x] += s1s[threadIdx.x + s];
        }
        __syncthreads();
    }
    if (threadIdx.x == 0)
        out[0] = 0.5f * (s0s[0] + s1s[0]) / (float)BN;
}

extern "C" void kernel_launch(void* const* d_in, const int* in_sizes, int n_in,
                              void* d_out, int out_size, void* d_ws, size_t ws_size,
                              hipStream_t stream) {
    const float* pred = (const float*)d_in[0];  // [B,N,3]
    const float* gt   = (const float*)d_in[1];  // [B,N,3]
    const float* P    = (const float*)d_in[2];  // [B,3,4]

    // Workspace layout (~1.3 MB total):
    float*    Apk  = (float*)d_ws;          // BN*4 floats
    float*    Bpk  = Apk + (size_t)BN * 4;  // BN*4 floats
    unsigned* gCol = (unsigned*)(Bpk + (size_t)BN * 4);  // BN uints (f32 bits)
    float*    gRow = (float*)(gCol + BN);                // BN floats

    pc_pack_kernel<<<BN / 256, 256, 0, stream>>>(pred, gt, P, Apk, Bpk, gCol);
    pc_chamfer_kernel<<<BB * (NN / 256), 256, 0, stream>>>(Apk, Bpk, gCol, gRow);
    pc_reduce_kernel<<<1, 256, 0, stream>>>(gCol, gRow, (float*)d_out);
}